// ChamferLoss_18210661335546
// MI455X (gfx1250) — compile-verified
//
#include <hip/hip_runtime.h>
#include <hip/hip_bf16.h>
#include <math.h>

// ---------------------------------------------------------------------------
// Chamfer soft-min loss, fused, for MI455X (gfx1250, wave32).
//
// dist tile (16x16) per V_WMMA_F32_16X16X4_F32:
//   A (16 refs x K4)  = (rx, ry, rz, r^2 [+ penalty])      -> a-frag v2f
//   B (K4 x 16 qrys)  = (-2qx, -2qy, -2qz, 1)^T            -> b-frag v2f
//   C                 = |q|^2 broadcast (per-lane scalar, N = lane&15)
//   D[m][n] = |q_n - r_m|^2 (+ penalty_m)
//
// C/D layout: lane group 0-15: N=lane, VGPR v = row M=v; lanes 16-31: M=v+8.
// Each lane owns one query column and 8 reference rows per tile, so the online
// softmax over references is purely per-lane. Two tiles per iteration; one
// branchless rescale per 16 elements (tile-pair min), exp2-domain softmax.
// ---------------------------------------------------------------------------

typedef float v2f __attribute__((ext_vector_type(2)));
typedef float v8f __attribute__((ext_vector_type(8)));

#define EPS_F     1e-6f
#define PENALTY_F 10.0f
#define NPTS      4096
#define NB        8
// logit in log2 domain: L = -(1/tau)*log2(e) * d = -14.426950408889634 * d
#define L2SCALE   (-14.426950408889634f)

// ---------------------------------------------------------------- init accum
__global__ void ch_init_kernel(float* accum) {
    if (threadIdx.x < 4) accum[threadIdx.x] = 0.0f;
}

// ------------------------------------------------- per-batch target min/max
__global__ __launch_bounds__(256) void ch_stats_kernel(const float* __restrict__ tgt,
                                                       float* __restrict__ cs) {
    const int b   = blockIdx.x;
    const int tid = threadIdx.x;
    const float* tb = tgt + (size_t)b * NPTS * 3;

    float mnx =  INFINITY, mny =  INFINITY, mnz =  INFINITY;
    float mxx = -INFINITY, mxy = -INFINITY, mxz = -INFINITY;
    for (int i = tid; i < NPTS; i += 256) {
        float x = tb[i * 3 + 0], y = tb[i * 3 + 1], z = tb[i * 3 + 2];
        mnx = fminf(mnx, x); mny = fminf(mny, y); mnz = fminf(mnz, z);
        mxx = fmaxf(mxx, x); mxy = fmaxf(mxy, y); mxz = fmaxf(mxz, z);
    }
    __shared__ float red[6][256];
    red[0][tid] = mnx; red[1][tid] = mny; red[2][tid] = mnz;
    red[3][tid] = mxx; red[4][tid] = mxy; red[5][tid] = mxz;
    for (int off = 128; off > 0; off >>= 1) {
        __syncthreads();
        if (tid < off) {
            red[0][tid] = fminf(red[0][tid], red[0][tid + off]);
            red[1][tid] = fminf(red[1][tid], red[1][tid + off]);
            red[2][tid] = fminf(red[2][tid], red[2][tid + off]);
            red[3][tid] = fmaxf(red[3][tid], red[3][tid + off]);
            red[4][tid] = fmaxf(red[4][tid], red[4][tid + off]);
            red[5][tid] = fmaxf(red[5][tid], red[5][tid + off]);
        }
    }
    __syncthreads();
    if (tid == 0) {
        float cx = 0.5f * (red[3][0] + red[0][0]);
        float cy = 0.5f * (red[4][0] + red[1][0]);
        float cz = 0.5f * (red[5][0] + red[2][0]);
        float sc = fmaxf(fmaxf(red[3][0] - red[0][0], red[4][0] - red[1][0]),
                         red[5][0] - red[2][0]) + EPS_F;
        cs[b * 4 + 0] = cx; cs[b * 4 + 1] = cy;
        cs[b * 4 + 2] = cz; cs[b * 4 + 3] = sc;
    }
}

// ------------------------------ normalize + pack float4 refs, mask sums
__global__ __launch_bounds__(256) void ch_pack_kernel(
    const float* __restrict__ src, const float* __restrict__ tgt,
    const float* __restrict__ ms,  const float* __restrict__ mt,
    const float* __restrict__ cs,
    float4* __restrict__ s4, float4* __restrict__ t4,
    float* __restrict__ s2p, float* __restrict__ accum) {

    const int idx = blockIdx.x * 256 + threadIdx.x;   // 0 .. B*NPTS-1
    const int b   = idx >> 12;
    const float4 c = ((const float4*)cs)[b];
    const float invs = 1.0f / c.w;

    float sx = (src[idx * 3 + 0] - c.x) * invs;
    float sy = (src[idx * 3 + 1] - c.y) * invs;
    float sz = (src[idx * 3 + 2] - c.z) * invs;
    float s2 = sx * sx + sy * sy + sz * sz;
    float wms = ms[idx];
    s4[idx]  = make_float4(sx, sy, sz, s2 + (1.0f - wms) * PENALTY_F);
    s2p[idx] = s2;

    float tx = (tgt[idx * 3 + 0] - c.x) * invs;
    float ty = (tgt[idx * 3 + 1] - c.y) * invs;
    float tz = (tgt[idx * 3 + 2] - c.z) * invs;
    float t2 = tx * tx + ty * ty + tz * tz;
    t4[idx] = make_float4(tx, ty, tz, t2);

    float wmt = mt[idx];
    float a0 = wms, a1 = wmt;
    #pragma unroll
    for (int off = 1; off < 32; off <<= 1) {
        a0 += __shfl_xor(a0, off);
        a1 += __shfl_xor(a1, off);
    }
    if ((threadIdx.x & 31) == 0) {
        atomicAdd(&accum[1], a0);
        atomicAdd(&accum[3], a1);
    }
}

// --------------------------------------------------- fused WMMA soft-min
// One workgroup = 8 waves x 16 queries = 128 queries; loops all 4096 refs.
__global__ __launch_bounds__(256) void ch_softmin_kernel(
    const float4* __restrict__ ref4,   // [B*NPTS] (x,y,z, r^2[+pen])
    const float4* __restrict__ q4,     // [B*NPTS] query positions (.w unused)
    const float*  __restrict__ q2b,    // plain |q|^2, strided
    int q2stride,
    const float*  __restrict__ qw,     // per-query loss weight
    float* __restrict__ accum, int numSlot) {

    __shared__ float4 lds[1024];                 // 16 KB staging (1024 refs)

    const int b    = blockIdx.x >> 5;            // 32 blocks per batch
    const int slab = blockIdx.x & 31;            // 128-query slab
    const int lane = threadIdx.x & 31;
    const int wave = threadIdx.x >> 5;
    const int col  = lane & 15;
    const int half = lane >> 4;                  // 0: (x,y)/K01  1: (z,w)/K23
    const int qi   = b * NPTS + slab * 128 + wave * 16 + col;

    // B fragment: lanes 0-15 -> (K0,K1)=(-2qx,-2qy); lanes 16-31 -> (K2,K3)=(-2qz,1)
    const float4 qp = q4[qi];
    v2f bfrag;
    bfrag[0] = (lane < 16) ? (-2.0f * qp.x) : (-2.0f * qp.z);
    bfrag[1] = (lane < 16) ? (-2.0f * qp.y) : 1.0f;
    const float cini = q2b[(size_t)qi * q2stride];
    v8f cm;
    #pragma unroll
    for (int v = 0; v < 8; ++v) cm[v] = cini;

    // online softmax state: m = running max of L (log2 logit), s = sum 2^(L-m),
    // t = sum d * 2^(L-m)
    float m = -INFINITY, s = 0.0f, t = 0.0f;

    const float4* refb = ref4 + (size_t)b * NPTS;

    for (int chunk = 0; chunk < 4; ++chunk) {
        for (int i = threadIdx.x; i < 1024; i += 256)
            lds[i] = refb[chunk * 1024 + i];
        __syncthreads();

        // software-pipelined A-fragment loads (2 tiles ahead)
        const float2* lp = reinterpret_cast<const float2*>(lds);
        float2 a0 = lp[(0 * 16 + col) * 2 + half];
        float2 a1 = lp[(1 * 16 + col) * 2 + half];

        for (int tt = 0; tt < 64; tt += 2) {
            const int nx = (tt + 2) & 63;        // wraps; dummy reload on last iter
            float2 n0 = lp[((nx + 0) * 16 + col) * 2 + half];
            float2 n1 = lp[((nx + 1) * 16 + col) * 2 + half];

            v2f af0; af0[0] = a0.x; af0[1] = a0.y;
            v2f af1; af1[0] = a1.x; af1[1] = a1.y;

            v8f d0 = __builtin_amdgcn_wmma_f32_16x16x4_f32(
                false, af0, false, bfrag, (short)0, cm, false, false);
            v8f d1 = __builtin_amdgcn_wmma_f32_16x16x4_f32(
                false, af1, false, bfrag, (short)0, cm, false, false);

            float dv[16];
            #pragma unroll
            for (int v = 0; v < 8; ++v) {
                dv[v]     = fmaxf(d0[v], 0.0f);
                dv[v + 8] = fmaxf(d1[v], 0.0f);
            }
            // tile-pair min distance -> max logit
            float mn01 = fminf(dv[0], dv[1]),   mn23 = fminf(dv[2], dv[3]);
            float mn45 = fminf(dv[4], dv[5]),   mn67 = fminf(dv[6], dv[7]);
            float mn89 = fminf(dv[8], dv[9]),   mnab = fminf(dv[10], dv[11]);
            float mncd = fminf(dv[12], dv[13]), mnef = fminf(dv[14], dv[15]);
            float mnA  = fminf(fminf(mn01, mn23), fminf(mn45, mn67));
            float mnB  = fminf(fminf(mn89, mnab), fminf(mncd, mnef));
            float Lmax = L2SCALE * fminf(mnA, mnB);

            // one branchless rescale per 16 elements
            float nm = fmaxf(m, Lmax);
            float r  = __builtin_amdgcn_exp2f(m - nm);
            s *= r;
            t *= r;
            m  = nm;

            #pragma unroll
            for (int v = 0; v < 16; ++v) {
                float e = __builtin_amdgcn_exp2f(L2SCALE * dv[v] - m);
                s += e;
                t += dv[v] * e;
            }

            a0 = n0;
            a1 = n1;
        }
        __syncthreads();
    }

    // merge the two half-wave partials (rows 0-7 with rows 8-15 of each column)
    float mo = __shfl_xor(m, 16);
    float so = __shfl_xor(s, 16);
    float to = __shfl_xor(t, 16);
    float mm = fmaxf(m, mo);
    float c1 = __builtin_amdgcn_exp2f(m - mm);
    float c2 = __builtin_amdgcn_exp2f(mo - mm);
    s = s * c1 + so * c2;
    t = t * c1 + to * c2;

    float contrib = (lane < 16) ? (qw[qi] * (t / s)) : 0.0f;
    #pragma unroll
    for (int off = 1; off < 32; off <<= 1) contrib += __shfl_xor(contrib, off);
    if (lane == 0) atomicAdd(&accum[numSlot], contrib);
}

// ---------------------------------------------------------------- finalize
__global__ void ch_final_kernel(const float* __restrict__ accum,
                                float* __restrict__ out) {
    out[0] = accum[0] / (accum[1] + EPS_F) + accum[2] / (accum[3] + EPS_F);
}

// ---------------------------------------------------------------------------
extern "C" void kernel_launch(void* const* d_in, const int* in_sizes, int n_in,
                              void* d_out, int out_size, void* d_ws, size_t ws_size,
                              hipStream_t stream) {
    (void)in_sizes; (void)n_in; (void)out_size; (void)ws_size;

    const float* src = (const float*)d_in[0];   // [B,N,3]
    const float* tgt = (const float*)d_in[1];   // [B,M,3]
    const float* ms  = (const float*)d_in[2];   // [B,N]
    const float* mt  = (const float*)d_in[3];   // [B,M]
    float* out = (float*)d_out;

    float* ws = (float*)d_ws;
    const size_t nBP = (size_t)NB * NPTS;       // 32768
    float4* t4   = (float4*)(ws);               // 4*nBP floats
    float4* s4   = (float4*)(ws + 4 * nBP);     // 4*nBP floats
    float*  s2p  = ws + 8 * nBP;                // nBP floats
    float*  cs   = ws + 9 * nBP;                // 4*NB floats
    float*  accum = cs + 4 * NB;                // 4 floats

    ch_init_kernel<<<1, 32, 0, stream>>>(accum);
    ch_stats_kernel<<<NB, 256, 0, stream>>>(tgt, cs);
    ch_pack_kernel<<<(int)(nBP / 256), 256, 0, stream>>>(src, tgt, ms, mt, cs,
                                                         s4, t4, s2p, accum);
    // source -> target : refs = targets (plain t^2), queries = sources (plain s^2)
    ch_softmin_kernel<<<NB * 32, 256, 0, stream>>>(t4, s4, s2p, 1, ms, accum, 0);
    // target -> source : refs = sources (s^2 + penalty), queries = targets
    ch_softmin_kernel<<<NB * 32, 256, 0, stream>>>(s4, t4, ((const float*)t4) + 3, 4,
                                                   mt, accum, 2);
    ch_final_kernel<<<1, 1, 0, stream>>>(accum, out);
}